// MultiScaleLocalEncoder_81870666596704
// MI455X (gfx1250) — compile-verified
//
#include <hip/hip_runtime.h>
#include <cstdint>
#include <cstddef>

typedef __attribute__((ext_vector_type(2))) float v2f;
typedef __attribute__((ext_vector_type(8))) float v8f;

namespace {

constexpr int kB = 4;
constexpr int kN = 8192;
constexpr int kM = 1024;
constexpr int kD = 32;
constexpr float kEps = 1e-5f;

// ---------------- FPS: one block (32 waves) per batch ----------------
__global__ void fps_kernel(const float* __restrict__ coords, int* __restrict__ anchor_idx)
{
    const int b = blockIdx.x;
    const int tid = threadIdx.x;                 // 0..1023, each owns 8 points
    const float* cb = coords + (size_t)b * kN * 3;

    float px[8], py[8], pz[8], dd[8];
#pragma unroll
    for (int i = 0; i < 8; ++i) {
        int j = tid + i * 1024;
        px[i] = cb[j * 3 + 0];
        py[i] = cb[j * 3 + 1];
        pz[i] = cb[j * 3 + 2];
        dd[i] = 1e10f;
    }

    __shared__ float s_last[3];
    __shared__ float s_bd[32];
    __shared__ int   s_bi[32];

    if (tid == 0) {
        s_last[0] = cb[0]; s_last[1] = cb[1]; s_last[2] = cb[2];
        anchor_idx[b * kM] = 0;
    }
    const int lane = tid & 31;
    const int wv = tid >> 5;

    for (int s = 1; s < kM; ++s) {
        __syncthreads();
        const float lx = s_last[0], ly = s_last[1], lz = s_last[2];
        float bd = -1.0f; int bi = 0x7fffffff;
#pragma unroll
        for (int i = 0; i < 8; ++i) {
            float dx = px[i] - lx, dy = py[i] - ly, dz = pz[i] - lz;
            float d = dx * dx + dy * dy + dz * dz;
            dd[i] = fminf(dd[i], d);
            int j = tid + i * 1024;
            if (dd[i] > bd) { bd = dd[i]; bi = j; }     // ascending j keeps first max
        }
#pragma unroll
        for (int off = 16; off > 0; off >>= 1) {        // wave32 argmax, first-index tie-break
            float od = __shfl_xor(bd, off, 32);
            int   oi = __shfl_xor(bi, off, 32);
            if (od > bd || (od == bd && oi < bi)) { bd = od; bi = oi; }
        }
        if (lane == 0) { s_bd[wv] = bd; s_bi[wv] = bi; }
        __syncthreads();
        if (wv == 0) {
            bd = s_bd[lane]; bi = s_bi[lane];
#pragma unroll
            for (int off = 16; off > 0; off >>= 1) {
                float od = __shfl_xor(bd, off, 32);
                int   oi = __shfl_xor(bi, off, 32);
                if (od > bd || (od == bd && oi < bi)) { bd = od; bi = oi; }
            }
            if (lane == 0) {
                anchor_idx[b * kM + s] = bi;
                s_last[0] = cb[bi * 3 + 0];
                s_last[1] = cb[bi * 3 + 1];
                s_last[2] = cb[bi * 3 + 2];
            }
        }
    }
}

__global__ void write_anchors_kernel(const float* __restrict__ coords,
                                     const int* __restrict__ anchor_idx,
                                     float* __restrict__ anchors,
                                     float* __restrict__ out)
{
    int t = blockIdx.x * blockDim.x + threadIdx.x;
    if (t >= kB * kM) return;
    int b = t >> 10;
    int pid = anchor_idx[t];
    const float* cp = coords + ((size_t)b * kN + pid) * 3;
    float x = cp[0], y = cp[1], z = cp[2];
    anchors[t * 3 + 0] = x; anchors[t * 3 + 1] = y; anchors[t * 3 + 2] = z;
    out[t * 3 + 0] = x;     out[t * 3 + 1] = y;     out[t * 3 + 2] = z;
}

// ------------- Ball query: one wave per anchor, all 3 radii in one scan -------------
__global__ void ball_query_kernel(const float* __restrict__ coords,
                                  const float* __restrict__ anchors,
                                  int* __restrict__ idx1,
                                  int* __restrict__ idx2,
                                  int* __restrict__ idx3)
{
    const int lane = threadIdx.x & 31;
    const int bm = blockIdx.x * (blockDim.x >> 5) + (threadIdx.x >> 5);
    if (bm >= kB * kM) return;
    const int b = bm >> 10;
    const float ax = anchors[bm * 3 + 0], ay = anchors[bm * 3 + 1], az = anchors[bm * 3 + 2];
    const float* cb = coords + (size_t)b * kN * 3;
    const float r1s = 0.1f * 0.1f, r2s = 0.2f * 0.2f, r3s = 0.4f * 0.4f;
    int c1 = 0, c2 = 0, c3 = 0;
    int f1 = 0, f2 = 0, f3 = 0;
    const unsigned lt = (1u << lane) - 1u;
    for (int base = 0; base < kN; base += 32) {
        int j = base + lane;
        float dx = cb[j * 3 + 0] - ax, dy = cb[j * 3 + 1] - ay, dz = cb[j * 3 + 2] - az;
        float d2 = dx * dx + dy * dy + dz * dz;
        bool p1 = d2 <= r1s, p2 = d2 <= r2s, p3 = d2 <= r3s;
        unsigned m1 = (unsigned)__ballot(p1);
        unsigned m2 = (unsigned)__ballot(p2);
        unsigned m3 = (unsigned)__ballot(p3);
        if (c1 == 0 && m1) f1 = base + __ffs(m1) - 1;
        if (c2 == 0 && m2) f2 = base + __ffs(m2) - 1;
        if (c3 == 0 && m3) f3 = base + __ffs(m3) - 1;
        if (p1) { int pos = c1 + __popc(m1 & lt); if (pos < 16) idx1[bm * 16 + pos] = j; }
        if (p2) { int pos = c2 + __popc(m2 & lt); if (pos < 32) idx2[bm * 32 + pos] = j; }
        if (p3) { int pos = c3 + __popc(m3 & lt); if (pos < 64) idx3[bm * 64 + pos] = j; }
        c1 = min(16, c1 + __popc(m1));
        c2 = min(32, c2 + __popc(m2));
        c3 = min(64, c3 + __popc(m3));
        if (c1 >= 16 && c2 >= 32 && c3 >= 64) break;
    }
    // pad with first found index (anchor itself always qualifies, but be safe)
    for (int s = c1 + lane; s < 16; s += 32) idx1[bm * 16 + s] = f1;
    for (int s = c2 + lane; s < 32; s += 32) idx2[bm * 32 + s] = f2;
    for (int s = c3 + lane; s < 64; s += 32) idx3[bm * 64 + s] = f3;
}

// ------------- Conv (1x1) with fused gather / BN+ReLU consume, fp32 WMMA -------------
// GEMM: out[Cout, P] = W[Cout, Cin] * act[Cin, P], tiled 16x16 with K-steps of 4
// using V_WMMA_F32_16X16X4_F32. Two column tiles are processed per iteration with
// independent accumulator chains (shared A fragment) to give the matrix pipe ILP.
// Per-channel sum/sumsq are accumulated per-lane and reduced cross-lane once.
template<int CINP, int CINR, int COUT, bool GATHER, int KVAL>
__global__ void conv_bn_kernel(const float* __restrict__ W,
                               const float* __restrict__ Xprev,
                               const float* __restrict__ scl,
                               const float* __restrict__ sft,
                               const int* __restrict__ idx,
                               const float* __restrict__ coords,
                               const float* __restrict__ feats,
                               const float* __restrict__ anchors,
                               float* __restrict__ Z,
                               float* __restrict__ gsum,
                               float* __restrict__ gsq)
{
    static_assert(CINP % 4 == 0 && COUT % 16 == 0, "tile shapes");
    constexpr int NK = CINP / 4;
    constexpr int NROW = COUT / 16;
    constexpr int NT = 2;                          // column tiles in flight per wave
    const int lane = threadIdx.x & 31;
    const int gw = blockIdx.x * (blockDim.x >> 5) + (threadIdx.x >> 5);
    const int row_tile = gw % NROW;
    const int colgrp = gw / NROW;
    const int half = lane >> 4;       // K-pair select within A/B fragments
    const int l16 = lane & 15;
    const int row = row_tile * 16 + l16;

    // Preload A fragments (weights): lane l<16 holds K={0,1}, lane l>=16 holds
    // K={2,3} of each 4-wide K step (ISA 32-bit A layout). For the consume path
    // also preload the previous layer's BN scale/shift (loop-invariant).
    float aw0[NK], aw1[NK];
    float sc0[NK], sc1[NK], sh0[NK], sh1[NK];
#pragma unroll
    for (int kk = 0; kk < NK; ++kk) {
        int ch0 = kk * 4 + 2 * half;
        aw0[kk] = (ch0 < CINR) ? W[row * CINR + ch0] : 0.f;
        aw1[kk] = (ch0 + 1 < CINR) ? W[row * CINR + ch0 + 1] : 0.f;
        if constexpr (!GATHER) {
            sc0[kk] = scl[ch0];     sh0[kk] = sft[ch0];
            sc1[kk] = scl[ch0 + 1]; sh1[kk] = sft[ch0 + 1];
        } else {
            sc0[kk] = sc1[kk] = sh0[kk] = sh1[kk] = 0.f;
        }
    }

    float s_acc[8], q_acc[8];
#pragma unroll
    for (int i = 0; i < 8; ++i) { s_acc[i] = 0.f; q_acc[i] = 0.f; }

    const int ct0 = colgrp * 64;                   // 64 column tiles per wave
    for (int ctb = ct0; ctb < ct0 + 64; ctb += NT) {
        int p[NT];
        float rl0[NT], rl1[NT], rl2[NT], f0[NT];
        const float* fsrc[NT];
#pragma unroll
        for (int t = 0; t < NT; ++t) {
            p[t] = (ctb + t) * 16 + l16;
            rl0[t] = rl1[t] = rl2[t] = f0[t] = 0.f;
            fsrc[t] = nullptr;
            if constexpr (GATHER) {
                int bm = p[t] / KVAL;              // KVAL is a power-of-two constant
                int b = bm >> 10;
                int pid = idx[p[t]];
                const float* cp = coords + ((size_t)b * kN + pid) * 3;
                rl0[t] = cp[0] - anchors[bm * 3 + 0];
                rl1[t] = cp[1] - anchors[bm * 3 + 1];
                rl2[t] = cp[2] - anchors[bm * 3 + 2];
                fsrc[t] = feats + ((size_t)b * kN + pid) * kD;
                f0[t] = fsrc[t][0];
            }
        }

        v8f acc[NT];
#pragma unroll
        for (int t = 0; t < NT; ++t) acc[t] = (v8f){};

#pragma unroll
        for (int kk = 0; kk < NK; ++kk) {
            const int ch0 = kk * 4 + 2 * half;     // per-lane channel base
            v2f afrag; afrag.x = aw0[kk]; afrag.y = aw1[kk];
#pragma unroll
            for (int t = 0; t < NT; ++t) {
                float x0, x1;
                if constexpr (GATHER) {
                    if (kk == 0) {
                        // channels: half0 -> {rel0, rel1}; half1 -> {rel2, feat[0]}
                        x0 = half ? rl2[t] : rl0[t];
                        x1 = half ? f0[t] : rl1[t];
                    } else if (kk == NK - 1) {
                        // channels: half0 -> {f29, f30}; half1 -> {f31, pad 0}
                        x0 = fsrc[t][ch0 - 3];
                        float xt = fsrc[t][(ch0 - 2 <= kD - 1) ? (ch0 - 2) : (kD - 1)];
                        x1 = half ? 0.f : xt;
                    } else {
                        x0 = fsrc[t][ch0 - 3];
                        x1 = fsrc[t][ch0 - 2];
                    }
                } else {
                    const float* zp = Xprev + (size_t)p[t] * CINR + ch0;
                    x0 = fmaxf(0.f, zp[0] * sc0[kk] + sh0[kk]);   // fused BN + ReLU
                    x1 = fmaxf(0.f, zp[1] * sc1[kk] + sh1[kk]);
                }
                v2f bfrag; bfrag.x = x0; bfrag.y = x1;
                acc[t] = __builtin_amdgcn_wmma_f32_16x16x4_f32(
                    false, afrag, false, bfrag, (short)0, acc[t], false, false);
            }
        }

        // D tile: VGPR i, lane l -> row (i + 8*half), col l16. 8 contiguous channels.
#pragma unroll
        for (int t = 0; t < NT; ++t) {
            float* outp = Z + (size_t)p[t] * COUT + row_tile * 16 + half * 8;
#pragma unroll
            for (int i = 0; i < 8; ++i) outp[i] = acc[t][i];
#pragma unroll
            for (int i = 0; i < 8; ++i) {
                s_acc[i] += acc[t][i];
                q_acc[i] += acc[t][i] * acc[t][i];
            }
        }
    }

    // single cross-lane reduction over the 16 columns of each half-wave
#pragma unroll
    for (int i = 0; i < 8; ++i) {
        float v = s_acc[i];
        float q = q_acc[i];
#pragma unroll
        for (int off = 1; off < 16; off <<= 1) {
            v += __shfl_xor(v, off, 32);
            q += __shfl_xor(q, off, 32);
        }
        s_acc[i] = v; q_acc[i] = q;
    }
    if (l16 == 0) {  // lane 0 (rows 0..7) and lane 16 (rows 8..15)
        const int chb = row_tile * 16 + half * 8;
#pragma unroll
        for (int i = 0; i < 8; ++i) {
            atomicAdd(&gsum[chb + i], s_acc[i]);
            atomicAdd(&gsq[chb + i], q_acc[i]);
        }
    }
}

__global__ void zero_kernel(float* p, int n)
{
    int i = blockIdx.x * blockDim.x + threadIdx.x;
    if (i < n) p[i] = 0.f;
}

__global__ void bn_finalize_kernel(const float* __restrict__ gsum,
                                   const float* __restrict__ gsq,
                                   const float* __restrict__ gamma,
                                   const float* __restrict__ beta,
                                   int C, float invPn,
                                   float* __restrict__ scl, float* __restrict__ sft)
{
    int c = threadIdx.x;
    if (c >= C) return;
    float mean = gsum[c] * invPn;
    float var = gsq[c] * invPn - mean * mean;
    float s = gamma[c] * rsqrtf(var + kEps);
    scl[c] = s;
    sft[c] = beta[c] - mean * s;
}

template<int COUT>
__global__ void maxpool_kernel(const float* __restrict__ Z,
                               const float* __restrict__ scl,
                               const float* __restrict__ sft,
                               int K, int chan_off, float* __restrict__ out)
{
    int t = blockIdx.x * blockDim.x + threadIdx.x;
    if (t >= kB * kM * COUT) return;
    int c = t % COUT;
    int m = (t / COUT) & (kM - 1);
    int b = t / (COUT * kM);
    float s = scl[c], h = sft[c];
    const float* zp = Z + ((size_t)(b * kM + m) * K) * COUT + c;
    float mx = 0.f;                               // ReLU outputs are >= 0
    for (int k = 0; k < K; ++k)
        mx = fmaxf(mx, fmaxf(0.f, zp[(size_t)k * COUT] * s + h));
    out[12288 + ((size_t)(b * 320 + chan_off + c)) * kM + m] = mx;
}

template<int CINP, int CINR, int COUT, bool GATHER, int KVAL>
static void launch_conv(hipStream_t stream, const float* W, const float* Xprev,
                        const float* scl, const float* sft,
                        const int* idx, const float* coords, const float* feats,
                        const float* anchors, int P,
                        float* Z, float* gsum, float* gsq)
{
    int waves = (COUT / 16) * (P / 16 / 64);   // P is a multiple of 1024 everywhere
    int blocks = waves / 8;                    // 8 waves (256 threads) per block
    conv_bn_kernel<CINP, CINR, COUT, GATHER, KVAL><<<blocks, 256, 0, stream>>>(
        W, Xprev, scl, sft, idx, coords, feats, anchors, Z, gsum, gsq);
}

template<int C1, int C2, int C3, int KVAL>
static void run_scale(hipStream_t stream,
                      const float* coords, const float* feats, const float* anchors,
                      const int* idx, int chan_off,
                      const float* const* prm,     // w0,g0,b0, w1,g1,b1, w2,g2,b2
                      float* zA, float* zB, float* stats, float* scl, float* sft,
                      float* out)
{
    const int P = kB * kM * KVAL;
    float* gsum = stats;
    float* gsq = stats + 128;
    const float invPn = 1.0f / (float)P;

    zero_kernel<<<1, 256, 0, stream>>>(stats, 256);
    launch_conv<36, 35, C1, true, KVAL>(stream, prm[0], nullptr, nullptr, nullptr,
                                        idx, coords, feats, anchors, P, zA, gsum, gsq);
    bn_finalize_kernel<<<1, 128, 0, stream>>>(gsum, gsq, prm[1], prm[2], C1, invPn, scl, sft);

    zero_kernel<<<1, 256, 0, stream>>>(stats, 256);
    launch_conv<C1, C1, C2, false, KVAL>(stream, prm[3], zA, scl, sft,
                                         nullptr, nullptr, nullptr, nullptr, P, zB, gsum, gsq);
    bn_finalize_kernel<<<1, 128, 0, stream>>>(gsum, gsq, prm[4], prm[5], C2, invPn, scl, sft);

    zero_kernel<<<1, 256, 0, stream>>>(stats, 256);
    launch_conv<C2, C2, C3, false, KVAL>(stream, prm[6], zB, scl, sft,
                                         nullptr, nullptr, nullptr, nullptr, P, zA, gsum, gsq);
    bn_finalize_kernel<<<1, 128, 0, stream>>>(gsum, gsq, prm[7], prm[8], C3, invPn, scl, sft);

    maxpool_kernel<C3><<<(kB * kM * C3) / 256, 256, 0, stream>>>(zA, scl, sft, KVAL, chan_off, out);
}

} // namespace

extern "C" void kernel_launch(void* const* d_in, const int* in_sizes, int n_in,
                              void* d_out, int out_size, void* d_ws, size_t ws_size,
                              hipStream_t stream)
{
    (void)in_sizes; (void)n_in; (void)out_size; (void)ws_size;
    const float* coords = (const float*)d_in[0];
    const float* feats = (const float*)d_in[1];
    const float* const* prm = (const float* const*)(d_in + 2);   // 27 param tensors

    char* ws = (char*)d_ws;
    size_t off = 0;
    auto alloc = [&](size_t bytes) -> void* {
        void* p = ws + off;
        off = (off + bytes + 255) & ~(size_t)255;
        return p;
    };
    int* anchor_idx = (int*)alloc((size_t)kB * kM * sizeof(int));
    float* anchors = (float*)alloc((size_t)kB * kM * 3 * sizeof(float));
    int* idx1 = (int*)alloc((size_t)kB * kM * 16 * sizeof(int));
    int* idx2 = (int*)alloc((size_t)kB * kM * 32 * sizeof(int));
    int* idx3 = (int*)alloc((size_t)kB * kM * 64 * sizeof(int));
    float* stats = (float*)alloc(256 * sizeof(float));
    float* scl = (float*)alloc(128 * sizeof(float));
    float* sft = (float*)alloc(128 * sizeof(float));
    float* zA = (float*)alloc((size_t)kB * kM * 64 * 128 * sizeof(float));   // 134 MB
    float* zB = (float*)alloc((size_t)kB * kM * 64 * 128 * sizeof(float));   // 134 MB

    float* out = (float*)d_out;

    fps_kernel<<<kB, 1024, 0, stream>>>(coords, anchor_idx);
    write_anchors_kernel<<<(kB * kM) / 256, 256, 0, stream>>>(coords, anchor_idx, anchors, out);
    ball_query_kernel<<<(kB * kM) / 8, 256, 0, stream>>>(coords, anchors, idx1, idx2, idx3);

    run_scale<32, 32, 64, 16>(stream, coords, feats, anchors, idx1, 0,   prm + 0,  zA, zB, stats, scl, sft, out);
    run_scale<64, 64, 128, 32>(stream, coords, feats, anchors, idx2, 64,  prm + 9,  zA, zB, stats, scl, sft, out);
    run_scale<64, 96, 128, 64>(stream, coords, feats, anchors, idx3, 192, prm + 18, zA, zB, stats, scl, sft, out);
}